// APPNPMessageLayer_37022618092150
// MI455X (gfx1250) — compile-verified
//
#include <hip/hip_runtime.h>

#define N_NODES 100000
#define HIDDEN  128
#define ALPHA   0.1f
#define K_HOPS  10
#define LDS_STRIDE 132   // 128 + 4 pad: lanes hit distinct LDS banks (stride%64 = 4)

typedef __attribute__((ext_vector_type(2))) float v2f;
typedef __attribute__((ext_vector_type(8))) float v8f;

// ---------------- setup kernels ----------------

__global__ void init_deg_kernel(float* __restrict__ deg, int n) {
    int i = blockIdx.x * blockDim.x + threadIdx.x;
    if (i < n) deg[i] = 1.0f;                 // self-loop contributes 1
}

__global__ void zero4_kernel(float4* __restrict__ p, int n4) {
    int i = blockIdx.x * blockDim.x + threadIdx.x;
    if (i < n4) p[i] = make_float4(0.f, 0.f, 0.f, 0.f);
}

__global__ void accum_deg_kernel(const int* __restrict__ dst,
                                 float* __restrict__ deg, int E) {
    int e = blockIdx.x * blockDim.x + threadIdx.x;
    if (e < E) unsafeAtomicAdd(&deg[dst[e]], 1.0f);   // native global_atomic_add_f32
}

__global__ void rsqrt_kernel(float* __restrict__ deg, int n) {
    int i = blockIdx.x * blockDim.x + threadIdx.x;
    if (i < n) deg[i] = rsqrtf(deg[i]);       // deg >= 1 always (self-loop)
}

__global__ void norm_kernel(const int* __restrict__ src, const int* __restrict__ dst,
                            const float* __restrict__ dinv,
                            float* __restrict__ norm, int E) {
    int e = blockIdx.x * blockDim.x + threadIdx.x;
    if (e < E) norm[e] = dinv[src[e]] * dinv[dst[e]];
}

// ---------------- WMMA GEMM: h0 = x @ W^T + b ----------------
// Block: 256 threads = 8 waves. Block handles a 16-row strip of x; wave w
// produces the 16x16 output tile for columns [16w, 16w+16).
// D[M,N] = sum_K A[M,K]*B[K,N] with A = x rows, B[K,N] = W[j0+N, K].
__global__ __launch_bounds__(256) void gemm_wmma_kernel(
    const float* __restrict__ x, const float* __restrict__ W,
    const float* __restrict__ bias, float* __restrict__ h0) {

    __shared__ float xs[16 * LDS_STRIDE];

    const int tid = threadIdx.x;
    const int n0  = blockIdx.x * 16;

    // Stage 16x128 x-tile into LDS: each of 256 threads moves 8 floats (2x float4)
    {
        int row = tid >> 4;             // 0..15
        int col = (tid & 15) * 8;       // 0..120
        const float4* g = (const float4*)(x + (size_t)(n0 + row) * HIDDEN + col);
        float4* s = (float4*)(xs + row * LDS_STRIDE + col);
        s[0] = g[0];
        s[1] = g[1];
    }
    __syncthreads();

    const int wave = tid >> 5;          // 0..7 -> column tile
    const int lane = tid & 31;
    const int half = lane >> 4;         // 0: K pair {0,1}, 1: K pair {2,3}
    const int nIdx = lane & 15;         // M for A, N for B/C
    const int j0   = wave * 16;

    const float* wrow = W + (size_t)(j0 + nIdx) * HIDDEN;  // B column j0+nIdx = W row
    const float  bval = bias[j0 + nIdx];

    v8f c = {};
    #pragma unroll 8
    for (int kk = 0; kk < 32; ++kk) {
        const int kb = kk * 4 + half * 2;
        v2f a, b;
        // A 16x4 layout: lanes 0-15 hold (M=lane, K=0,1); lanes 16-31 (M=lane-16, K=2,3)
        a.x = xs[nIdx * LDS_STRIDE + kb];
        a.y = xs[nIdx * LDS_STRIDE + kb + 1];
        // B 4x16 layout mirrors: lanes 0-15 (K=0,1, N=lane); lanes 16-31 (K=2,3, N=lane-16)
        b.x = wrow[kb];
        b.y = wrow[kb + 1];
        c = __builtin_amdgcn_wmma_f32_16x16x4_f32(
                /*neg_a=*/false, a, /*neg_b=*/false, b,
                /*c_mod=*/(short)0, c, /*reuse_a=*/false, /*reuse_b=*/false);
    }

    // C/D layout: VGPR i -> lanes 0-15: (M=i, N=lane); lanes 16-31: (M=8+i, N=lane-16)
    #pragma unroll
    for (int i = 0; i < 8; ++i) {
        const int m = i + half * 8;
        h0[(size_t)(n0 + m) * HIDDEN + j0 + nIdx] = c[i] + bval;
    }
}

// ---------------- propagation: one wave per edge ----------------
// lane L moves features [4L, 4L+4): 512B coalesced gather + 4 native f32 atomics.
__global__ __launch_bounds__(256) void scatter_kernel(
    const float* __restrict__ h, const float* __restrict__ norm,
    const int* __restrict__ src, const int* __restrict__ dst,
    float* __restrict__ agg, int E) {

    const int wid  = (blockIdx.x * 256 + threadIdx.x) >> 5;  // edge id
    if (wid >= E) return;
    const int lane = threadIdx.x & 31;

    const int   s  = src[wid];
    const int   d  = dst[wid];
    const float nm = norm[wid];

    float4 v = ((const float4*)(h + (size_t)s * HIDDEN))[lane];
    float* a = agg + (size_t)d * HIDDEN + lane * 4;
    unsafeAtomicAdd(a + 0, nm * v.x);
    unsafeAtomicAdd(a + 1, nm * v.y);
    unsafeAtomicAdd(a + 2, nm * v.z);
    unsafeAtomicAdd(a + 3, nm * v.w);
}

// h_new = (1-a)*(agg + dinv^2 * h_old) + a*h0 ; re-zero agg for next hop.
__global__ __launch_bounds__(256) void combine_kernel(
    float* __restrict__ agg, const float* __restrict__ hold,
    const float* __restrict__ h0, const float* __restrict__ dinv,
    float* __restrict__ hnew, int n4) {

    const int idx = blockIdx.x * 256 + threadIdx.x;   // float4 index, node = idx/32
    if (idx >= n4) return;
    const int   node = idx >> 5;
    const float di   = dinv[node];
    const float sw   = di * di;                       // self-loop norm

    float4 a  = ((const float4*)agg)[idx];
    float4 ho = ((const float4*)hold)[idx];
    float4 hz = ((const float4*)h0)[idx];

    float4 r;
    r.x = (1.0f - ALPHA) * (a.x + sw * ho.x) + ALPHA * hz.x;
    r.y = (1.0f - ALPHA) * (a.y + sw * ho.y) + ALPHA * hz.y;
    r.z = (1.0f - ALPHA) * (a.z + sw * ho.z) + ALPHA * hz.z;
    r.w = (1.0f - ALPHA) * (a.w + sw * ho.w) + ALPHA * hz.w;

    ((float4*)hnew)[idx] = r;
    ((float4*)agg)[idx]  = make_float4(0.f, 0.f, 0.f, 0.f);
}

// ---------------- launcher ----------------

extern "C" void kernel_launch(void* const* d_in, const int* in_sizes, int n_in,
                              void* d_out, int out_size, void* d_ws, size_t ws_size,
                              hipStream_t stream) {
    const float* x    = (const float*)d_in[0];
    const int*   ei   = (const int*)d_in[1];       // [2, E] flat: row0 = src, row1 = dst
    const float* W    = (const float*)d_in[2];
    const float* bias = (const float*)d_in[3];
    float*       out  = (float*)d_out;

    const int E = in_sizes[1] / 2;
    const int* src = ei;
    const int* dst = ei + E;

    // workspace layout (floats)
    float* ws   = (float*)d_ws;
    float* h0   = ws;                                   // N*128
    float* hbuf = h0   + (size_t)N_NODES * HIDDEN;      // N*128
    float* agg  = hbuf + (size_t)N_NODES * HIDDEN;      // N*128
    float* deg  = agg  + (size_t)N_NODES * HIDDEN;      // N  (becomes dinv in place)
    float* norm = deg  + N_NODES;                       // E

    const int n4 = N_NODES * (HIDDEN / 4);              // 3.2M float4s

    init_deg_kernel<<<(N_NODES + 255) / 256, 256, 0, stream>>>(deg, N_NODES);
    zero4_kernel<<<(n4 + 255) / 256, 256, 0, stream>>>((float4*)agg, n4);
    accum_deg_kernel<<<(E + 255) / 256, 256, 0, stream>>>(dst, deg, E);
    rsqrt_kernel<<<(N_NODES + 255) / 256, 256, 0, stream>>>(deg, N_NODES);
    norm_kernel<<<(E + 255) / 256, 256, 0, stream>>>(src, dst, deg, norm, E);

    gemm_wmma_kernel<<<N_NODES / 16, 256, 0, stream>>>(x, W, bias, h0);

    const float* hin = h0;                              // hop 1 reads h0 directly
    for (int hop = 0; hop < K_HOPS; ++hop) {
        scatter_kernel<<<(E + 7) / 8, 256, 0, stream>>>(hin, norm, src, dst, agg, E);
        float* hout = (hop == K_HOPS - 1) ? out : hbuf;
        combine_kernel<<<(n4 + 255) / 256, 256, 0, stream>>>(agg, hin, h0, deg, hout, n4);
        hin = hbuf;
    }
}